// GNNRLAgent_46256797778298
// MI455X (gfx1250) — compile-verified
//
#include <hip/hip_runtime.h>
#include <hip/hip_bf16.h>

#define B_  16
#define N_  50000
#define E_  800000
#define K_  1000
#define C_  8
#define BN_ (B_ * N_)
#define P_  1002          // K_ + 2 policy rows per batch

typedef float v2f __attribute__((ext_vector_type(2)));
typedef float v8f __attribute__((ext_vector_type(8)));

// ---------------------------------------------------------------------------
// Pass 1: in-degree over col (float atomics; L2 RMW, no return -> STOREcnt path)
// grid = (E_/256, B_)
__global__ void deg_kernel(const long long* __restrict__ ei, float* __restrict__ deg) {
    const int b = blockIdx.y;
    const int i = blockIdx.x * blockDim.x + threadIdx.x;          // < E_
    const long long c = ei[(size_t)b * 2 * E_ + E_ + i];          // col value
    atomicAdd(&deg[b * N_ + (int)c], 1.0f);
    if (b == 0 && i == 0) atomicAdd(&deg[0], 1.0f);               // the single (0,0) self-loop
}

// Pass 2: dinv = deg>0 ? rsqrt(deg) : 0   (in place)
__global__ void dinv_kernel(float* __restrict__ deg) {
    const int j = blockIdx.x * blockDim.x + threadIdx.x;          // < BN_
    const float d = deg[j];
    deg[j] = (d > 0.0f) ? rsqrtf(d) : 0.0f;
}

// Pass 3: t[col] += dinv[row] * x[row]   (scalar aggregation; dinv[col] applied later)
// row stream is read exactly once -> non-temporal so the col stream stays hot in L2.
__global__ void agg_kernel(const long long* __restrict__ ei, const float* __restrict__ x,
                           const float* __restrict__ dinv, float* __restrict__ t) {
    const int b = blockIdx.y;
    const int i = blockIdx.x * blockDim.x + threadIdx.x;
    const size_t base = (size_t)b * 2 * E_;
    const int row = (int)__builtin_nontemporal_load(&ei[base + i]);
    const int col = (int)ei[base + E_ + i];
    const int r = b * N_ + row;
    atomicAdd(&t[b * N_ + col], dinv[r] * x[r]);
    if (b == 0 && i == 0) atomicAdd(&t[0], dinv[0] * x[0]);       // self-loop contribution
}

// Pass 4: vsum[b] = sum_i dinv[b,i]*t[b,i]   (block reduce + one atomic per block)
// grid = (ceil(N/256), B_)
__global__ void vsum_kernel(const float* __restrict__ dinv, const float* __restrict__ t,
                            float* __restrict__ vsum) {
    __shared__ float sdata[256];
    const int b = blockIdx.y;
    const int local = blockIdx.x * blockDim.x + threadIdx.x;
    float s = 0.0f;
    if (local < N_) {
        const int j = b * N_ + local;
        s = dinv[j] * t[j];
    }
    sdata[threadIdx.x] = s;
    __syncthreads();
    for (int off = 128; off > 0; off >>= 1) {
        if ((int)threadIdx.x < off) sdata[threadIdx.x] += sdata[threadIdx.x + off];
        __syncthreads();
    }
    if (threadIdx.x == 0) atomicAdd(&vsum[b], sdata[0]);
}

// Pass 5: policy head via V_WMMA_F32_16X16X4_F32.
// One wave (32 lanes) computes a 16-row tile of policy_in[16032,8] @ Wp[8] as
// two accumulating K=4 WMMAs. B broadcasts Wp over all 16 columns, so every
// column of D equals P@Wp; D is read back via the documented C/D layout.
__global__ __launch_bounds__(32) void policy_kernel(
    const float* __restrict__ dinv, const float* __restrict__ t,
    const long long* __restrict__ defi,
    const float* __restrict__ Wg, const float* __restrict__ bg,
    const float* __restrict__ wait_e, const float* __restrict__ term_e,
    const float* __restrict__ Wp, const float* __restrict__ bp,
    float* __restrict__ out)
{
    const int T    = blockIdx.x;          // 0..1001 (1002 tiles * 16 rows = 16032)
    const int lane = threadIdx.x;
    const int m    = lane & 15;
    const int hi   = lane >> 4;           // 0: K=0,1 half  1: K=2,3 half
    const int r    = T * 16 + m;          // flattened policy row
    const int b    = r / P_;
    const int k    = r - b * P_;

    float s = 0.0f;
    int mode;
    if (k < K_) {
        const int d = (int)defi[b * K_ + k];
        const int j = b * N_ + d;
        s = dinv[j] * t[j];
        mode = 0;                          // defense row: s*Wg[c] + bg[c]
    } else {
        mode = (k == K_) ? 1 : 2;          // wait / term embedding rows
    }
    auto elem = [&](int c) -> float {
        if (mode == 0) return s * Wg[c] + bg[c];
        return (mode == 1) ? wait_e[c] : term_e[c];
    };

    // A 16x4 f32 layout: lane<16 -> {K=0,K=1} in VGPR0/1 ; lane>=16 -> {K=2,K=3}
    const int c0 = hi ? 2 : 0;
    v2f A0 = { elem(c0 + 0), elem(c0 + 1) };      // K chunk 0 (c = 0..3)
    v2f A1 = { elem(c0 + 4), elem(c0 + 5) };      // K chunk 1 (c = 4..7)
    // B 4x16: row K broadcast across columns -> value depends only on K
    v2f B0 = { Wp[c0 + 0], Wp[c0 + 1] };
    v2f B1 = { Wp[c0 + 4], Wp[c0 + 5] };

    v8f acc = {};
    acc = __builtin_amdgcn_wmma_f32_16x16x4_f32(false, A0, false, B0, (short)0, acc, false, false);
    acc = __builtin_amdgcn_wmma_f32_16x16x4_f32(false, A1, false, B1, (short)0, acc, false, false);

    // D layout: lane 0 holds rows 0..7 (any column), lane 16 holds rows 8..15
    const float bpv = bp[0];
    if (m == 0) {
        const int rowbase = T * 16 + hi * 8;
#pragma unroll
        for (int v = 0; v < 8; ++v) out[rowbase + v] = acc[v] + bpv;
    }
}

// Pass 6: value head (16 scalars)
__global__ void value_kernel(const float* __restrict__ vsum,
                             const float* __restrict__ Wg, const float* __restrict__ bg,
                             const float* __restrict__ Wv, const float* __restrict__ bv,
                             float* __restrict__ out) {
    const int b = threadIdx.x;
    if (b < B_) {
        float dWgWv = 0.0f, dBgWv = 0.0f;
#pragma unroll
        for (int c = 0; c < C_; ++c) { dWgWv += Wg[c] * Wv[c]; dBgWv += bg[c] * Wv[c]; }
        out[B_ * P_ + b] = vsum[b] * (1.0f / N_) * dWgWv + dBgWv + bv[0];
    }
}

extern "C" void kernel_launch(void* const* d_in, const int* in_sizes, int n_in,
                              void* d_out, int out_size, void* d_ws, size_t ws_size,
                              hipStream_t stream) {
    const float*     x      = (const float*)d_in[0];
    const long long* ei     = (const long long*)d_in[1];
    const long long* defi   = (const long long*)d_in[2];
    const float*     Wg     = (const float*)d_in[3];
    const float*     bg     = (const float*)d_in[4];
    const float*     wait_e = (const float*)d_in[5];
    const float*     term_e = (const float*)d_in[6];
    const float*     Wp     = (const float*)d_in[7];
    const float*     bp     = (const float*)d_in[8];
    const float*     Wv     = (const float*)d_in[9];
    const float*     bv     = (const float*)d_in[10];
    float* out = (float*)d_out;

    // workspace: [deg->dinv : BN] [t : BN] [vsum : 16]
    float* deg  = (float*)d_ws;
    float* t    = deg + BN_;
    float* vsum = t + BN_;
    hipMemsetAsync(d_ws, 0, (size_t)(2 * BN_ + 16) * sizeof(float), stream);

    const dim3 eg(E_ / 256, B_);                 // 3125 x 16 blocks, 1 thread/edge
    deg_kernel<<<eg, 256, 0, stream>>>(ei, deg);
    dinv_kernel<<<BN_ / 256, 256, 0, stream>>>(deg);          // 3125 blocks exact
    agg_kernel<<<eg, 256, 0, stream>>>(ei, x, deg, t);
    const dim3 vg((N_ + 255) / 256, B_);
    vsum_kernel<<<vg, 256, 0, stream>>>(deg, t, vsum);
    policy_kernel<<<(B_ * P_) / 16, 32, 0, stream>>>(deg, t, defi, Wg, bg,
                                                     wait_e, term_e, Wp, bp, out);
    value_kernel<<<1, 32, 0, stream>>>(vsum, Wg, bg, Wv, bv, out);
}